// SSD_5669356831967
// MI455X (gfx1250) — compile-verified
//
#include <hip/hip_runtime.h>
#include <stdint.h>

#define NUM_CLASSES 81
#define TOP_K 200
#define CONF_THRESH 0.01f
#define NMS_THRESH 0.45f
#define BATCH 64
#define NUM_PRIORS 8732
#define PAD 208      // 13 * 16, padded NMS matrix dimension
#define NTILE 13

// ---------------------------------------------------------------------------
// Kernel 1: per-(batch, prior) softmax statistics (max logit, 1/sum(exp)).
// Stages a 64-prior x 81-class tile through LDS so global reads are fully
// coalesced; conf is never written back (saves 362 MB vs a transposed plan).
// ---------------------------------------------------------------------------
__global__ __launch_bounds__(256)
void ssd_softmax_stats(const float* __restrict__ conf,
                       float* __restrict__ mx, float* __restrict__ ise) {
    __shared__ float tile[64 * NUM_CLASSES];   // 20736 B
    const int b  = blockIdx.y;
    const int p0 = blockIdx.x * 64;
    const int np = min(64, NUM_PRIORS - p0);
    const float* src = conf + ((size_t)b * NUM_PRIORS + p0) * NUM_CLASSES;
    const int total = np * NUM_CLASSES;
    for (int i = threadIdx.x; i < total; i += blockDim.x) tile[i] = src[i];
    __syncthreads();
    if ((int)threadIdx.x < np) {
        const float* row = tile + threadIdx.x * NUM_CLASSES;
        float m = row[0];
        for (int c = 1; c < NUM_CLASSES; ++c) m = fmaxf(m, row[c]);
        float s = 0.0f;
        for (int c = 0; c < NUM_CLASSES; ++c) s += expf(row[c] - m);
        const size_t p = (size_t)b * NUM_PRIORS + (p0 + threadIdx.x);
        mx[p]  = m;
        ise[p] = 1.0f / s;       // reciprocal once here, multiply 81x later
    }
}

// ---------------------------------------------------------------------------
// Kernel 2: box decode (elementwise, 64*8732 threads).
// ---------------------------------------------------------------------------
__global__ __launch_bounds__(256)
void ssd_decode(const float* __restrict__ loc, const float* __restrict__ prior,
                float* __restrict__ dec) {
    const int idx = blockIdx.x * blockDim.x + threadIdx.x;
    if (idx >= BATCH * NUM_PRIORS) return;
    const int p = idx % NUM_PRIORS;
    const float* L = loc + (size_t)idx * 10;
    const float px = prior[p * 4 + 0], py = prior[p * 4 + 1];
    const float pw = prior[p * 4 + 2], ph = prior[p * 4 + 3];
    const float cx = px + L[0] * 0.1f * pw;
    const float cy = py + L[1] * 0.1f * ph;
    const float w  = pw * expf(L[2] * 0.2f);
    const float h  = ph * expf(L[3] * 0.2f);
    const float g1x = fmaxf(pw * (expf(L[4] * 0.2f) - 0.1f), 0.0f);
    const float g1y = fmaxf(ph * (expf(L[5] * 0.2f) - 0.1f), 0.0f);
    const float g2x = fmaxf(pw * (expf(L[6] * 0.2f) - 0.1f), 0.0f);
    const float g2y = fmaxf(ph * (expf(L[7] * 0.2f) - 0.1f), 0.0f);
    const float g3x = px + L[8] * 0.1f * pw;
    const float g3y = py + L[9] * 0.1f * ph;
    const float x1 = cx - w * 0.5f, y1 = cy - h * 0.5f;
    const float x2 = x1 + w,        y2 = y1 + h;
    float* D = dec + (size_t)idx * 10;
    D[0] = x1;  D[1] = y1;  D[2] = x2;  D[3] = y2;
    D[4] = g1x; D[5] = g1y; D[6] = g2x; D[7] = g2y; D[8] = g3x; D[9] = g3y;
}

// LDS union: the three members are live in disjoint phases.
union ScoreOver {
    float s[NUM_PRIORS];                 // phase 1-2: scores      (34928 B)
    unsigned char over[PAD * PAD];       // phase 4-5: overlap mask (43264 B)
    float outb[TOP_K * 11];              // phase 6: staged output  ( 8800 B)
};

typedef float v2f __attribute__((ext_vector_type(2)));
typedef float v8f __attribute__((ext_vector_type(8)));

// ---------------------------------------------------------------------------
// Kernel 3: one block (8 wave32) per (image, class) task.
//   1. on-the-fly softmax score + threshold into LDS
//   2. top-200: iterative argmax, register-cached slice maxima,
//      wave32 shfl_xor butterfly reduce (2 barriers per extraction)
//   3. gather top boxes into LDS (SoA)
//   4. 200x200 overlap mask; area-sum via V_WMMA_F32_16X16X4_F32 rank-2 GEMM;
//      division-free IoU threshold (denominator provably > 0)
//   5. sequential greedy suppression (matches the reference scan exactly)
//   6. stage rows in LDS, stream out coalesced (class 0 forced to zero)
// ---------------------------------------------------------------------------
__global__ __launch_bounds__(256)
void ssd_nms(const float* __restrict__ conf, const float* __restrict__ mx,
             const float* __restrict__ ise, const float* __restrict__ dec,
             float* __restrict__ out) {
    __shared__ float bx1[PAD], by1[PAD], bx2[PAD], by2[PAD], bar[PAD];
    __shared__ float topv[TOP_K];
    __shared__ int   topi[TOP_K];
    __shared__ float rv[8];
    __shared__ int   ri[8];
    __shared__ float gwv;
    __shared__ int   gwi;
    __shared__ unsigned char keep[PAD];
    __shared__ ScoreOver u;

    const int b    = blockIdx.x / NUM_CLASSES;
    const int c    = blockIdx.x % NUM_CLASSES;
    const int tid  = threadIdx.x;
    const int lane = tid & 31;          // wave32
    const int wave = tid >> 5;
    const size_t bp = (size_t)b * NUM_PRIORS;
    float* O = out + (size_t)(b * NUM_CLASSES + c) * TOP_K * 11;

    if (c == 0) {   // reference zeroes class 0 entirely
        for (int q = tid; q < TOP_K * 11; q += 256) O[q] = 0.0f;
        return;
    }

    // ---- Phase 1: scores ---------------------------------------------------
    for (int p = tid; p < NUM_PRIORS; p += 256) {
        if (p + 256 < NUM_PRIORS)
            __builtin_prefetch(&conf[(bp + p + 256) * NUM_CLASSES + c], 0, 0);
        const float v = expf(conf[(bp + p) * NUM_CLASSES + c] - mx[bp + p]) * ise[bp + p];
        u.s[p] = (v > CONF_THRESH) ? v : 0.0f;
    }
    __syncthreads();

    // ---- Phase 2: top-200 --------------------------------------------------
    float cv = -1.0f; int ci = 0x7fffffff;       // register-cached slice argmax
    for (int p = tid; p < NUM_PRIORS; p += 256) {
        const float v = u.s[p];
        if (v > cv) { cv = v; ci = p; }          // ascending scan => lowest index on ties
    }
    for (int k = 0; k < TOP_K; ++k) {
        float v = cv; int i = ci;
#pragma unroll
        for (int off = 16; off > 0; off >>= 1) { // wave32 butterfly argmax
            const float ov = __shfl_xor(v, off);
            const int   oi = __shfl_xor(i, off);
            if (ov > v || (ov == v && oi < i)) { v = ov; i = oi; }
        }
        if (lane == 0) { rv[wave] = v; ri[wave] = i; }
        __syncthreads();
        if (wave == 0) {                         // reduce the 8 wave winners
            float v2 = (lane < 8) ? rv[lane] : -2.0f;
            int   i2 = (lane < 8) ? ri[lane] : 0x7fffffff;
#pragma unroll
            for (int off = 4; off > 0; off >>= 1) {
                const float ov = __shfl_xor(v2, off);
                const int   oi = __shfl_xor(i2, off);
                if (ov > v2 || (ov == v2 && oi < i2)) { v2 = ov; i2 = oi; }
            }
            if (lane == 0) {
                gwv = v2; gwi = i2;
                topv[k] = v2; topi[k] = i2;
                u.s[i2] = -1.0f;                 // retire the winner
            }
        }
        __syncthreads();
        const int widx = gwi;
        if ((widx & 255) == tid) {               // only the owner rescans its slice
            cv = -1.0f; ci = 0x7fffffff;
            for (int p = tid; p < NUM_PRIORS; p += 256) {
                const float vv = u.s[p];
                if (vv > cv) { cv = vv; ci = p; }
            }
        }
    }

    // ---- Phase 3: gather top boxes (SoA in LDS) ---------------------------
    for (int k = tid; k < PAD; k += 256) {
        if (k < TOP_K) {
            const float* D = dec + (bp + topi[k]) * 10;
            const float x1 = D[0], y1 = D[1], x2 = D[2], y2 = D[3];
            bx1[k] = x1; by1[k] = y1; bx2[k] = x2; by2[k] = y2;
            bar[k] = (x2 - x1) * (y2 - y1);
        } else {
            bx1[k] = 0.0f; by1[k] = 0.0f; bx2[k] = 0.0f; by2[k] = 0.0f;
            bar[k] = 0.0f;
        }
        keep[k] = 1;
    }
    __syncthreads();   // retires u.s before the union is reused as u.over

    // ---- Phase 4: overlap mask --------------------------------------------
    // denom = area_i + area_j - inter + 1e-12 > 0 always (areas > 0,
    // inter <= min(area_i, area_j)), so iou > t  <=>  inter > t * denom.
#if defined(__gfx1250__) && __has_builtin(__builtin_amdgcn_wmma_f32_16x16x4_f32)
    for (int t = wave; t < NTILE * NTILE; t += 8) {
        const int rb = (t / NTILE) * 16;
        const int cb = (t % NTILE) * 16;
        // A(16x4) = [areas_row | 1 | 0 | 0],  B(4x16) = [1 ; areas_col ; 0 ; 0]
        // => D[m][n] = areas_row[m] + areas_col[n]   (rank-2 f32 WMMA)
        v2f a, bb;
        if (lane < 16) {
            a.x  = bar[rb + lane]; a.y  = 1.0f;      // K=0, K=1
            bb.x = 1.0f;           bb.y = bar[cb + lane];
        } else {
            a.x = 0.0f; a.y = 0.0f;                  // K=2, K=3 rows are zero
            bb.x = 0.0f; bb.y = 0.0f;
        }
        v8f csum = {};
        csum = __builtin_amdgcn_wmma_f32_16x16x4_f32(
            false, a, false, bb, (short)0, csum, false, false);
        const int n     = cb + (lane & 15);
        const int mbase = rb + ((lane < 16) ? 0 : 8);
        const float cx1 = bx1[n], cy1 = by1[n], cx2 = bx2[n], cy2 = by2[n];
#pragma unroll
        for (int r = 0; r < 8; ++r) {
            const int m = mbase + r;
            const float ix1 = fmaxf(bx1[m], cx1);
            const float iy1 = fmaxf(by1[m], cy1);
            const float ix2 = fminf(bx2[m], cx2);
            const float iy2 = fminf(by2[m], cy2);
            const float inter = fmaxf(ix2 - ix1, 0.0f) * fmaxf(iy2 - iy1, 0.0f);
            u.over[m * PAD + n] =
                (inter > NMS_THRESH * (csum[r] - inter + 1e-12f)) ? 1 : 0;
        }
    }
#else
    for (int e = tid; e < TOP_K * TOP_K; e += 256) {
        const int m = e / TOP_K, n = e % TOP_K;
        const float ix1 = fmaxf(bx1[m], bx1[n]);
        const float iy1 = fmaxf(by1[m], by1[n]);
        const float ix2 = fminf(bx2[m], bx2[n]);
        const float iy2 = fminf(by2[m], by2[n]);
        const float inter = fmaxf(ix2 - ix1, 0.0f) * fmaxf(iy2 - iy1, 0.0f);
        u.over[m * PAD + n] =
            (inter > NMS_THRESH * (bar[m] + bar[n] - inter + 1e-12f)) ? 1 : 0;
    }
#endif
    __syncthreads();

    // ---- Phase 5: greedy suppression (sequential in i, matches scan) ------
    for (int i = 0; i < TOP_K; ++i) {
        if (keep[i]) {                            // uniform branch (LDS scalar)
            for (int j = tid; j < TOP_K; j += 256) {
                if (j > i && u.over[i * PAD + j]) keep[j] = 0;
            }
        }
        __syncthreads();
    }

    // ---- Phase 6: stage rows in LDS, stream out coalesced ------------------
    for (int k = tid; k < TOP_K; k += 256) {
        float* row = &u.outb[k * 11];
        const bool valid = keep[k] && (topv[k] > CONF_THRESH);
        if (valid) {
            const float* D = dec + (bp + topi[k]) * 10;
            row[0] = topv[k];
#pragma unroll
            for (int q = 0; q < 10; ++q) row[1 + q] = D[q];
        } else {
#pragma unroll
            for (int q = 0; q < 11; ++q) row[q] = 0.0f;
        }
    }
    __syncthreads();
    for (int q = tid; q < TOP_K * 11; q += 256) O[q] = u.outb[q];
}

// ---------------------------------------------------------------------------
extern "C" void kernel_launch(void* const* d_in, const int* in_sizes, int n_in,
                              void* d_out, int out_size, void* d_ws, size_t ws_size,
                              hipStream_t stream) {
    const float* loc   = (const float*)d_in[0];   // (64, 8732, 10)
    const float* conf  = (const float*)d_in[1];   // (64, 8732, 81)
    const float* prior = (const float*)d_in[2];   // (8732, 4)
    float* out = (float*)d_out;                   // (64, 81, 200, 11)

    // Workspace layout: mx | ise | dec   (~25.6 MB)
    float* mx  = (float*)d_ws;
    float* ise = mx + (size_t)BATCH * NUM_PRIORS;
    float* dec = ise + (size_t)BATCH * NUM_PRIORS;

    {
        dim3 grid((NUM_PRIORS + 63) / 64, BATCH);
        ssd_softmax_stats<<<grid, 256, 0, stream>>>(conf, mx, ise);
    }
    {
        const int n = BATCH * NUM_PRIORS;
        ssd_decode<<<(n + 255) / 256, 256, 0, stream>>>(loc, prior, dec);
    }
    {
        ssd_nms<<<BATCH * NUM_CLASSES, 256, 0, stream>>>(conf, mx, ise, dec, out);
    }
}